// GCN_Weights_13718125543763
// MI455X (gfx1250) — compile-verified
//
#include <hip/hip_runtime.h>
#include <hip/hip_bf16.h>
#include <math.h>

// ---------------------------------------------------------------------------
// GCN pipeline for gfx1250 (MI455X, wave32, WMMA).
// Dense GEMMs use the native fp32 tensor op V_WMMA_F32_16X16X4_F32.
// Irregular edge phase uses flat atomics (L2-resident working set).
// ---------------------------------------------------------------------------

typedef __attribute__((ext_vector_type(2))) float v2f;
typedef __attribute__((ext_vector_type(8))) float v8f;

#define FEAT    14
#define HDIM    64
#define HID     32
#define EMB_OUT 52

__device__ __forceinline__ float gelu_exact(float x) {
    // torch-default exact GELU: 0.5*x*(1+erf(x/sqrt(2)))
    return 0.5f * x * (1.0f + erff(x * 0.70710678118654752440f));
}

// -------------------------------- fills ------------------------------------
__global__ void k_fill(float* __restrict__ p, float v, int n) {
    int i = blockIdx.x * blockDim.x + threadIdx.x;
    if (i < n) p[i] = v;
}

// --------------------- edge-weight scatter (softmax comp) ------------------
__global__ void k_scatter_ew(float* __restrict__ ew, const int* __restrict__ mask,
                             int m, const float* __restrict__ msg_w, int comp, int E) {
    int i = blockIdx.x * blockDim.x + threadIdx.x;
    if (i >= m) return;
    float w0 = msg_w[0], w1 = msg_w[1], w2 = msg_w[2];
    float mx = fmaxf(w0, fmaxf(w1, w2));
    float e0 = expf(w0 - mx), e1 = expf(w1 - mx), e2 = expf(w2 - mx);
    float inv = 1.0f / (e0 + e1 + e2);
    float sp = (comp == 0 ? e0 : (comp == 1 ? e1 : e2)) * inv;
    int idx = mask[i];
    if (idx >= 0 && idx < E) ew[idx] = sp;
}

// ------------------------- node embedding (-> 64 dim) ----------------------
__global__ void k_embed(const float* __restrict__ x, const float* __restrict__ emb_W,
                        const float* __restrict__ emb_b, float* __restrict__ h, int N) {
    int t = blockIdx.x * blockDim.x + threadIdx.x;
    if (t >= N * HDIM) return;
    int n = t >> 6, c = t & 63;
    float v;
    if (c < EMB_OUT) {
        v = x[n * FEAT + 0] * emb_W[c] + x[n * FEAT + 1] * emb_W[EMB_OUT + c] + emb_b[c];
    } else {
        v = x[n * FEAT + (c - EMB_OUT + 2)];
    }
    h[t] = v;
}

// ----------------------- degree accumulation / rsqrt -----------------------
__global__ void k_deg(const int* __restrict__ col, const float* __restrict__ ew,
                      float* __restrict__ deg, int E) {
    int e = blockIdx.x * blockDim.x + threadIdx.x;
    if (e < E) atomicAdd(&deg[col[e]], ew[e]);
}

__global__ void k_dis(const float* __restrict__ deg, float* __restrict__ dis, int N) {
    int n = blockIdx.x * blockDim.x + threadIdx.x;
    if (n < N) dis[n] = rsqrtf(deg[n]);   // deg >= 1 due to self-loop weight 1
}

// --------------------- WMMA fp32 GEMM: C[Mx32] = A[MxK] * B[Kx32] ----------
// One wave per 16-row tile; two v8f accumulators cover the 32 output cols.
// A 16x4 fp32 tile: lanes 0-15 hold K={k,k+1}, lanes 16-31 hold K={k+2,k+3}.
// B 4x16 fp32 tile: per-lane column, VGPR0/1 = successive K in each lane-half.
__global__ void k_gemm_wmma32(const float* __restrict__ A, const float* __restrict__ B,
                              float* __restrict__ C, int M, int K) {
    int wave = (blockIdx.x * blockDim.x + threadIdx.x) >> 5;
    int lane = threadIdx.x & 31;
    int row0 = wave * 16;
    if (row0 >= M) return;                 // wave-uniform: EXEC stays all-ones
    int half = lane >> 4;
    int lr   = lane & 15;
    int arow = row0 + lr; if (arow >= M) arow = M - 1;   // clamp ragged tail
    v8f acc0 = {}; v8f acc1 = {};
    for (int k = 0; k < K; k += 4) {
        int ka = k + 2 * half;
        const float* ap = A + (long long)arow * K + ka;
        v2f a;  a.x  = ap[0];       a.y  = ap[1];
        const float* bp = B + (long long)ka * 32 + lr;
        v2f b0; b0.x = bp[0];       b0.y = bp[32];
        v2f b1; b1.x = bp[16];      b1.y = bp[48];
        acc0 = __builtin_amdgcn_wmma_f32_16x16x4_f32(false, a, false, b0,
                                                     (short)0, acc0, false, false);
        acc1 = __builtin_amdgcn_wmma_f32_16x16x4_f32(false, a, false, b1,
                                                     (short)0, acc1, false, false);
    }
#pragma unroll
    for (int j = 0; j < 8; ++j) {
        int r = row0 + j + 8 * half;       // C/D layout: VGPR j = row j (+8 hi half)
        if (r < M) {
            C[(long long)r * 32 + lr]      = acc0[j];
            C[(long long)r * 32 + 16 + lr] = acc1[j];
        }
    }
}

// -------- WMMA GEMM fused with bias+GELU (fc1): t = gelu(g*W + b), K=32 ----
__global__ void k_gemm_fc1(const float* __restrict__ A, const float* __restrict__ B,
                           const float* __restrict__ bias, float* __restrict__ C, int M) {
    int wave = (blockIdx.x * blockDim.x + threadIdx.x) >> 5;
    int lane = threadIdx.x & 31;
    int row0 = wave * 16;
    if (row0 >= M) return;
    int half = lane >> 4;
    int lr   = lane & 15;
    int arow = row0 + lr; if (arow >= M) arow = M - 1;
    float b_lo = bias[lr], b_hi = bias[16 + lr];
    v8f acc0 = {}; v8f acc1 = {};
    for (int k = 0; k < HID; k += 4) {
        int ka = k + 2 * half;
        const float* ap = A + (long long)arow * HID + ka;
        v2f a;  a.x  = ap[0];       a.y  = ap[1];
        const float* bp = B + (long long)ka * 32 + lr;
        v2f b0; b0.x = bp[0];       b0.y = bp[32];
        v2f b1; b1.x = bp[16];      b1.y = bp[48];
        acc0 = __builtin_amdgcn_wmma_f32_16x16x4_f32(false, a, false, b0,
                                                     (short)0, acc0, false, false);
        acc1 = __builtin_amdgcn_wmma_f32_16x16x4_f32(false, a, false, b1,
                                                     (short)0, acc1, false, false);
    }
#pragma unroll
    for (int j = 0; j < 8; ++j) {
        int r = row0 + j + 8 * half;
        if (r < M) {
            C[(long long)r * 32 + lr]      = gelu_exact(acc0[j] + b_lo);
            C[(long long)r * 32 + 16 + lr] = gelu_exact(acc1[j] + b_hi);
        }
    }
}

// -------------------- edge message scatter (atomic, L2-resident) -----------
__global__ void k_edge_msg(const int* __restrict__ row, const int* __restrict__ col,
                           const float* __restrict__ ew, const float* __restrict__ dis,
                           const float* __restrict__ hw, float* __restrict__ out,
                           unsigned int total) {
    unsigned int t = blockIdx.x * blockDim.x + threadIdx.x;
    if (t >= total) return;
    int e = (int)(t >> 5);
    int c = (int)(t & 31u);
    int r = row[e], d = col[e];
    float norm = dis[r] * ew[e] * dis[d];
    atomicAdd(&out[(long long)d * 32 + c], hw[(long long)r * 32 + c] * norm);
}

// ----- self-loop + bias + GELU + global_add_pool into per-graph buffer -----
__global__ void k_node_finish(const float* __restrict__ out, const float* __restrict__ hw,
                              const float* __restrict__ dis, const float* __restrict__ gcn_b,
                              const int* __restrict__ batch, float* __restrict__ g, int N) {
    int t = blockIdx.x * blockDim.x + threadIdx.x;
    if (t >= N * HID) return;
    int n = t >> 5, c = t & 31;
    float dn = dis[n];
    float v = out[t] + hw[t] * (dn * dn) + gcn_b[c];   // self-loop norm = dis[n]^2 * 1
    v = gelu_exact(v);
    atomicAdd(&g[(long long)batch[n] * HID + c], v);
}

// ------------------------------ fc2 (dot-32) -------------------------------
__global__ void k_fc2(const float* __restrict__ t, const float* __restrict__ W,
                      const float* __restrict__ b, float* __restrict__ logits, int G) {
    int gi = blockIdx.x * blockDim.x + threadIdx.x;
    if (gi >= G) return;
    float s = b[0];
#pragma unroll
    for (int c = 0; c < HID; ++c) s += t[(long long)gi * HID + c] * W[c];
    logits[gi] = s;
}

// ---------------------------------------------------------------------------
extern "C" void kernel_launch(void* const* d_in, const int* in_sizes, int n_in,
                              void* d_out, int out_size, void* d_ws, size_t ws_size,
                              hipStream_t stream) {
    const float* x       = (const float*)d_in[0];
    const int*   ei      = (const int*)  d_in[1];
    const int*   batch   = (const int*)  d_in[2];
    const int*   known   = (const int*)  d_in[3];
    const int*   unk     = (const int*)  d_in[4];
    const int*   obs     = (const int*)  d_in[5];
    const float* msg_w   = (const float*)d_in[6];
    const float* emb_W   = (const float*)d_in[7];
    const float* emb_b   = (const float*)d_in[8];
    const float* gcn_W   = (const float*)d_in[9];
    const float* gcn_b   = (const float*)d_in[10];
    const float* fc1_W   = (const float*)d_in[11];
    const float* fc1_b   = (const float*)d_in[12];
    const float* fc2_W   = (const float*)d_in[13];
    const float* fc2_b   = (const float*)d_in[14];

    const int N = in_sizes[0] / FEAT;
    const int E = in_sizes[1] / 2;
    const int M = in_sizes[3];
    const int G = out_size;

    const int* e_row = ei;        // edge_index[0]
    const int* e_col = ei + E;    // edge_index[1]

    // workspace layout (floats)
    float* ws   = (float*)d_ws;
    float* ew   = ws;                         // E
    float* h    = ew  + E;                    // N*64
    float* hw   = h   + (long long)N * HDIM;  // N*32
    float* deg  = hw  + (long long)N * HID;   // N
    float* dis  = deg + N;                    // N
    float* outb = dis + N;                    // N*32
    float* gbuf = outb + (long long)N * HID;  // G*32
    float* tbuf = gbuf + (long long)G * HID;  // G*32

    const int B = 256;
    auto cdiv = [](long long a, long long b) { return (int)((a + b - 1) / b); };

    // 1. init buffers
    k_fill<<<cdiv(E, B), B, 0, stream>>>(ew, 1.0f, E);
    k_fill<<<cdiv(N, B), B, 0, stream>>>(deg, 1.0f, N);                 // self-loops
    k_fill<<<cdiv((long long)N * HID, B), B, 0, stream>>>(outb, 0.0f, N * HID);
    k_fill<<<cdiv((long long)G * HID, B), B, 0, stream>>>(gbuf, 0.0f, G * HID);

    // 2. edge-weight scatters (sequential launches preserve last-write-wins order)
    k_scatter_ew<<<cdiv(M, B), B, 0, stream>>>(ew, known, M, msg_w, 0, E);
    k_scatter_ew<<<cdiv(M, B), B, 0, stream>>>(ew, unk,   M, msg_w, 1, E);
    k_scatter_ew<<<cdiv(M, B), B, 0, stream>>>(ew, obs,   M, msg_w, 2, E);

    // 3. embed first two features, pass through the rest -> h[N,64]
    k_embed<<<cdiv((long long)N * HDIM, B), B, 0, stream>>>(x, emb_W, emb_b, h, N);

    // 4. symmetric normalization
    k_deg<<<cdiv(E, B), B, 0, stream>>>(e_col, ew, deg, E);
    k_dis<<<cdiv(N, B), B, 0, stream>>>(deg, dis, N);

    // 5. hw = h @ gcn_W  (WMMA fp32, one wave per 16-row tile, 8 waves/block)
    {
        int tiles = (N + 15) / 16;
        int waves_per_block = B / 32;
        k_gemm_wmma32<<<cdiv(tiles, waves_per_block), B, 0, stream>>>(h, gcn_W, hw, N, HDIM);
    }

    // 6. edge scatter: out[col] += hw[row] * norm   (L2-resident atomics)
    {
        unsigned int total = (unsigned int)E * 32u;
        k_edge_msg<<<cdiv((long long)total, B), B, 0, stream>>>(e_row, e_col, ew, dis, hw,
                                                                outb, total);
    }

    // 7. self-loop + bias + GELU + global_add_pool -> gbuf[G,32]
    k_node_finish<<<cdiv((long long)N * HID, B), B, 0, stream>>>(outb, hw, dis, gcn_b,
                                                                 batch, gbuf, N);

    // 8. tbuf = gelu(gbuf @ fc1_W + fc1_b)  (WMMA fp32, fused epilogue)
    {
        int tiles = (G + 15) / 16;
        int waves_per_block = B / 32;
        k_gemm_fc1<<<cdiv(tiles, waves_per_block), B, 0, stream>>>(gbuf, fc1_W, fc1_b, tbuf, G);
    }

    // 9. logits = tbuf @ fc2_W + fc2_b
    k_fc2<<<cdiv(G, B), B, 0, stream>>>(tbuf, fc2_W, fc2_b, (float*)d_out, G);
}